// Qwen2Attention_60636348285673
// MI455X (gfx1250) — compile-verified
//
#include <hip/hip_runtime.h>
#include <hip/hip_bf16.h>

// ---------------- problem constants ----------------
#define BB    2
#define SS    2048
#define HID   2048
#define HQ    16
#define HKV   2
#define DD    128
#define NREP  8          // HQ / HKV
#define NEGV  (-1000000000.0f)
#define SCALEV 0.08838834764831845f   // 128^-0.5

typedef float v2f __attribute__((ext_vector_type(2)));
typedef float v8f __attribute__((ext_vector_type(8)));

static __device__ __forceinline__ v8f vzero8() {
    v8f z = {0.f,0.f,0.f,0.f,0.f,0.f,0.f,0.f};
    return z;
}

// D = A(16x4) * B(4x16) + C, fp32 WMMA (CDNA5)
static __device__ __forceinline__ v8f wmma_f32(v2f a, v2f b, v8f c) {
    return __builtin_amdgcn_wmma_f32_16x16x4_f32(false, a, false, b, (short)0, c, false, false);
}

// =====================================================================
// Kernel 1: fused QKV projection. X(4096x2048) * [Wq|Wk|Wv](2048x2560)+bias
// Output scattered to (B,H,S,D) layouts for Q(raw), K(raw), V.
// One 16x16 tile per wave; 4 waves per block. NT = 2560/16 = 160 tiles wide.
// =====================================================================
__global__ __launch_bounds__(128) void qkv_kernel(
    const float* __restrict__ X,
    const float* __restrict__ Wq, const float* __restrict__ bq,
    const float* __restrict__ Wk, const float* __restrict__ bk,
    const float* __restrict__ Wv, const float* __restrict__ bv,
    float* __restrict__ qraw, float* __restrict__ kraw, float* __restrict__ vbuf)
{
    const int lane = threadIdx.x & 31;
    const int wv   = threadIdx.x >> 5;
    const int tile = blockIdx.x * 4 + wv;
    const int NT   = 160;
    const int nt   = tile % NT;
    const int mt   = tile / NT;

    const int m  = lane & 15;          // A row / B col within tile
    const int kk = (lane >> 4) * 2;    // k sub-offset {0,2}

    const float* Arow = X + (size_t)(mt * 16 + m) * HID;
    const int ncol = nt * 16 + m;      // uniform per tile across lane halves

    // select weight matrix (tile-uniform condition; 2304 = 2048+256)
    const float* Wp; const float* bp; int ldw; int nloc;
    if (ncol < 2048)      { Wp = Wq; bp = bq; ldw = 2048; nloc = ncol;        }
    else if (ncol < 2304) { Wp = Wk; bp = bk; ldw = 256;  nloc = ncol - 2048; }
    else                  { Wp = Wv; bp = bv; ldw = 256;  nloc = ncol - 2304; }

    v8f acc = vzero8();
    for (int k0 = 0; k0 < HID; k0 += 4) {
        v2f a = *(const v2f*)(Arow + k0 + kk);
        v2f b;
        b.x = Wp[(size_t)(k0 + kk)     * ldw + nloc];
        b.y = Wp[(size_t)(k0 + kk + 1) * ldw + nloc];
        acc = wmma_f32(a, b, acc);
    }

    const float bias = bp[nloc];
    #pragma unroll
    for (int r = 0; r < 8; ++r) {
        const int mloc  = r + (lane >> 4) * 8;
        const int token = mt * 16 + mloc;
        const int b0 = token >> 11;          // batch
        const int s0 = token & (SS - 1);     // seq pos
        const float val = acc[r] + bias;
        if (ncol < 2048) {
            const int hh = ncol >> 7, d = ncol & 127;
            qraw[((size_t)((b0 * HQ + hh) * SS + s0)) * DD + d] = val;
        } else if (ncol < 2304) {
            const int n2 = ncol - 2048, hh = n2 >> 7, d = n2 & 127;
            kraw[((size_t)((b0 * HKV + hh) * SS + s0)) * DD + d] = val;
        } else {
            const int n2 = ncol - 2304, hh = n2 >> 7, d = n2 & 127;
            vbuf[((size_t)((b0 * HKV + hh) * SS + s0)) * DD + d] = val;
        }
    }
}

// =====================================================================
// Kernel 2: RoPE applied in place to Q and K: out = x*cos + rotate_half(x)*sin
// One thread per (row, d<64) pair. Q pairs first, then K pairs.
// =====================================================================
__global__ void rope_kernel(float* __restrict__ qraw, float* __restrict__ kraw,
                            const float* __restrict__ cosb, const float* __restrict__ sinb)
{
    const int QP = BB * HQ  * SS * (DD / 2);   // 4194304
    const int KP = BB * HKV * SS * (DD / 2);   //  524288
    const int idx = blockIdx.x * blockDim.x + threadIdx.x;
    if (idx < QP) {
        const int d = idx & 63;
        const int s = (idx >> 6) & (SS - 1);
        const int h = (idx >> 17) & (HQ - 1);
        const int b = idx >> 21;
        float* base = qraw + ((size_t)((b * HQ + h) * SS + s)) * DD;
        const float* cp = cosb + ((size_t)(b * SS + s)) * DD;
        const float* sp = sinb + ((size_t)(b * SS + s)) * DD;
        const float x1 = base[d], x2 = base[d + 64];
        base[d]      = x1 * cp[d]      - x2 * sp[d];
        base[d + 64] = x2 * cp[d + 64] + x1 * sp[d + 64];
    } else if (idx < QP + KP) {
        const int k = idx - QP;
        const int d = k & 63;
        const int s = (k >> 6) & (SS - 1);
        const int h = (k >> 17) & (HKV - 1);
        const int b = k >> 18;
        float* base = kraw + ((size_t)((b * HKV + h) * SS + s)) * DD;
        const float* cp = cosb + ((size_t)(b * SS + s)) * DD;
        const float* sp = sinb + ((size_t)(b * SS + s)) * DD;
        const float x1 = base[d], x2 = base[d + 64];
        base[d]      = x1 * cp[d]      - x2 * sp[d];
        base[d + 64] = x2 * cp[d + 64] + x1 * sp[d + 64];
    }
}

// =====================================================================
// Kernel 3: attention for one (b, h, 16-row q tile) per block (4 waves).
//   sweep1: scores = scale*Q K^T + causal mask (WMMA, waves stride kv tiles)
//           -> raw scores into attn_weights buffer (L2-hot), row-max tracked
//   sweep2: e = exp(s - max), row sums (LDS reduce)
//   sweep2b: normalize weights in place (final 512MB output)
//   sweep3: O = W @ V via WMMA; each wave owns 32 of 128 D columns
// =====================================================================
__global__ __launch_bounds__(128) void attn_kernel(
    const float* __restrict__ qraw, const float* __restrict__ kraw,
    const float* __restrict__ vbuf, float* __restrict__ attw,
    float* __restrict__ obuf)
{
    __shared__ float s_red[128 * 8];   // per-thread partial row maxima
    __shared__ float s_rowmax[16];
    __shared__ float s_sum[128];
    __shared__ float s_inv[16];

    const int lane = threadIdx.x & 31;
    const int wv   = threadIdx.x >> 5;

    const int bid = blockIdx.x;
    const int qt  = bid & 127;          // q tile (S/16 = 128)
    const int bh  = bid >> 7;
    const int h   = bh & (HQ - 1);
    const int b   = bh >> 4;
    const int kvh = h >> 3;             // h / NREP

    const float* Qt = qraw + ((size_t)((b * HQ  + h  ) * SS + qt * 16)) * DD;
    const float* Kb = kraw + ((size_t)((b * HKV + kvh) * SS)) * DD;
    const float* Vb = vbuf + ((size_t)((b * HKV + kvh) * SS)) * DD;
    float* Wout = attw + ((size_t)((b * HQ + h) * SS + qt * 16)) * SS;

    const int m  = lane & 15;
    const int kk = (lane >> 4) * 2;

    // preload Q tile A-operands (16x128 -> 32 k-steps of 4)
    v2f aq[32];
    #pragma unroll
    for (int t = 0; t < 32; ++t)
        aq[t] = *(const v2f*)(Qt + (size_t)m * DD + t * 4 + kk);

    float lmax[8];
    #pragma unroll
    for (int r = 0; r < 8; ++r) lmax[r] = -3.4e38f;

    // ---- sweep 1: scores ----
    for (int j = wv; j < 128; j += 4) {
        const float* Kt = Kb + (size_t)j * 16 * DD;
        v8f acc = vzero8();
        #pragma unroll
        for (int t = 0; t < 32; ++t) {
            v2f bk = *(const v2f*)(Kt + (size_t)m * DD + t * 4 + kk);
            acc = wmma_f32(aq[t], bk, acc);
        }
        const int kcol = j * 16 + m;    // C-layout: col = lane&15
        #pragma unroll
        for (int r = 0; r < 8; ++r) {
            const int mloc = r + (lane >> 4) * 8;
            const int qrow = qt * 16 + mloc;
            const float sv = acc[r] * SCALEV + ((kcol <= qrow) ? 0.0f : NEGV);
            Wout[(size_t)mloc * SS + kcol] = sv;
            lmax[r] = fmaxf(lmax[r], sv);
        }
    }
    #pragma unroll
    for (int r = 0; r < 8; ++r) s_red[threadIdx.x * 8 + r] = lmax[r];
    __threadfence();
    __syncthreads();

    // reduce row maxima: row mm lives in threads (wv*32 + (mm>>3)*16 + l), slot mm&7
    if (threadIdx.x < 16) {
        const int mm = threadIdx.x;
        float mx = -3.4e38f;
        for (int w = 0; w < 4; ++w)
            for (int l = 0; l < 16; ++l)
                mx = fmaxf(mx, s_red[(w * 32 + (mm >> 3) * 16 + l) * 8 + (mm & 7)]);
        s_rowmax[mm] = mx;
    }
    __syncthreads();

    // ---- sweep 2: exp + row sums ----
    const int row  = threadIdx.x >> 3;
    const int slot = threadIdx.x & 7;
    {
        const float mx = s_rowmax[row];
        float* wrow = Wout + (size_t)row * SS;
        float sum = 0.0f;
        for (int i = 0; i < 256; ++i) {
            const int c = slot + 8 * i;
            const float e = __expf(wrow[c] - mx);
            wrow[c] = e;
            sum += e;
        }
        s_sum[threadIdx.x] = sum;
    }
    __threadfence();
    __syncthreads();
    if (threadIdx.x < 16) {
        float s = 0.0f;
        for (int q = 0; q < 8; ++q) s += s_sum[threadIdx.x * 8 + q];
        s_inv[threadIdx.x] = 1.0f / s;
    }
    __syncthreads();

    // ---- sweep 2b: normalize (final attn_weights) ----
    {
        const float inv = s_inv[row];
        float* wrow = Wout + (size_t)row * SS;
        for (int i = 0; i < 256; ++i) {
            const int c = slot + 8 * i;
            wrow[c] *= inv;
        }
    }
    __threadfence();
    __syncthreads();

    // ---- sweep 3: O = W @ V ; wave wv owns D columns [wv*32, wv*32+32) ----
    const int nb0 = wv * 2;             // two 16-col blocks per wave
    v8f o0 = vzero8();
    v8f o1 = vzero8();
    for (int j = 0; j < 128; ++j) {
        const float* Vt = Vb + (size_t)j * 16 * DD;
        #pragma unroll
        for (int t = 0; t < 4; ++t) {
            v2f aw = *(const v2f*)(Wout + (size_t)m * SS + j * 16 + t * 4 + kk);
            const int kr = t * 4 + kk;
            v2f b0, b1;
            b0.x = Vt[(size_t)kr       * DD + nb0 * 16 + m];
            b0.y = Vt[(size_t)(kr + 1) * DD + nb0 * 16 + m];
            b1.x = Vt[(size_t)kr       * DD + (nb0 + 1) * 16 + m];
            b1.y = Vt[(size_t)(kr + 1) * DD + (nb0 + 1) * 16 + m];
            o0 = wmma_f32(aw, b0, o0);
            o1 = wmma_f32(aw, b1, o1);
        }
    }
    // store O in (B, S, HQ*D) layout for the output GEMM
    #pragma unroll
    for (int r = 0; r < 8; ++r) {
        const int mloc = r + (lane >> 4) * 8;
        const size_t rowbase = ((size_t)(b * SS + qt * 16 + mloc)) * (HQ * DD) + h * DD;
        obuf[rowbase + nb0 * 16 + m]       = o0[r];
        obuf[rowbase + (nb0 + 1) * 16 + m] = o1[r];
    }
}

// =====================================================================
// Kernel 4: output projection  O(4096x2048) @ Wo(2048x2048) -> attn_output
// =====================================================================
__global__ __launch_bounds__(128) void out_kernel(
    const float* __restrict__ Xo, const float* __restrict__ Wo,
    float* __restrict__ out)
{
    const int lane = threadIdx.x & 31;
    const int wv   = threadIdx.x >> 5;
    const int tile = blockIdx.x * 4 + wv;
    const int nt   = tile & 127;        // 2048/16 = 128
    const int mt   = tile >> 7;

    const int m  = lane & 15;
    const int kk = (lane >> 4) * 2;

    const float* Arow = Xo + (size_t)(mt * 16 + m) * HID;
    const int ncol = nt * 16 + m;

    v8f acc = vzero8();
    for (int k0 = 0; k0 < HID; k0 += 4) {
        v2f a = *(const v2f*)(Arow + k0 + kk);
        v2f b;
        b.x = Wo[(size_t)(k0 + kk)     * HID + ncol];
        b.y = Wo[(size_t)(k0 + kk + 1) * HID + ncol];
        acc = wmma_f32(a, b, acc);
    }
    #pragma unroll
    for (int r = 0; r < 8; ++r) {
        const int mloc = r + (lane >> 4) * 8;
        out[(size_t)(mt * 16 + mloc) * HID + ncol] = acc[r];
    }
}

// =====================================================================
extern "C" void kernel_launch(void* const* d_in, const int* in_sizes, int n_in,
                              void* d_out, int out_size, void* d_ws, size_t ws_size,
                              hipStream_t stream)
{
    (void)in_sizes; (void)n_in; (void)out_size; (void)ws_size;

    const float* hs   = (const float*)d_in[0];
    const float* cosb = (const float*)d_in[1];
    const float* sinb = (const float*)d_in[2];
    // d_in[3] = attn_mask (causal mask recomputed analytically)
    const float* Wq = (const float*)d_in[4];
    const float* bq = (const float*)d_in[5];
    const float* Wk = (const float*)d_in[6];
    const float* bk = (const float*)d_in[7];
    const float* Wv = (const float*)d_in[8];
    const float* bv = (const float*)d_in[9];
    const float* Wo = (const float*)d_in[10];

    float* attn_out = (float*)d_out;                             // B*S*HID
    float* attw     = attn_out + (size_t)BB * SS * HID;          // B*HQ*S*S

    // workspace layout (floats)
    float* qraw = (float*)d_ws;                                  //  8388608
    float* kraw = qraw + (size_t)BB * HQ  * SS * DD;             //  1048576
    float* vbuf = kraw + (size_t)BB * HKV * SS * DD;             //  1048576
    float* obuf = vbuf + (size_t)BB * HKV * SS * DD;             //  8388608

    // 1) QKV projection: (4096/16)*(2560/16) = 40960 tiles, 4 waves/block
    qkv_kernel<<<dim3(40960 / 4), dim3(128), 0, stream>>>(
        hs, Wq, bq, Wk, bk, Wv, bv, qraw, kraw, vbuf);

    // 2) RoPE in place on Q and K
    {
        const int total = BB * (HQ + HKV) * SS * (DD / 2);       // 4718592
        rope_kernel<<<dim3((total + 255) / 256), dim3(256), 0, stream>>>(
            qraw, kraw, cosb, sinb);
    }

    // 3) attention: B*HQ*(S/16) = 4096 blocks
    attn_kernel<<<dim3(BB * HQ * (SS / 16)), dim3(128), 0, stream>>>(
        qraw, kraw, vbuf, attw, obuf);

    // 4) output projection: (4096/16)*(2048/16) = 32768 tiles, 4 waves/block
    out_kernel<<<dim3(32768 / 4), dim3(128), 0, stream>>>(obuf, Wo, attn_out);
}